// TransformerGDInterleaved_88802743812162
// MI455X (gfx1250) — compile-verified
//
#include <hip/hip_runtime.h>
#include <math.h>

// Problem constants (from the reference)
#define B_      32
#define N_      1024
#define DIN_    144
#define EMB_    64
#define NC_     32
#define INSZ    16
#define NQ      16
#define MKV     (N_ - NQ)    // 1008 keys
#define NLAYERS 3

typedef __attribute__((ext_vector_type(2))) float v2f;
typedef __attribute__((ext_vector_type(8))) float v8f;

// Full-precision fp32 WMMA: D(16x16,f32) = A(16x4,f32) * B(4x16,f32) + C
__device__ __forceinline__ v8f wmma4(v2f a, v2f b, v8f c) {
  return __builtin_amdgcn_wmma_f32_16x16x4_f32(false, a, false, b, (short)0, c,
                                               false, false);
}
__device__ __forceinline__ v8f zero8() {
  v8f z = {0.f, 0.f, 0.f, 0.f, 0.f, 0.f, 0.f, 0.f};
  return z;
}

// Async global->LDS copy (CDNA5, tracked by ASYNCcnt).  Per-lane LDS
// destination address lets this perform the transpose scatter for free.
// LDS byte offset = low 32 bits of the generic address of a __shared__ object
// (the LDS aperture is a 4GB-aligned region, low bits are the LDS offset).
__device__ __forceinline__ void async_ld_b32(void* lds_dst, const void* gsrc) {
  unsigned dst = (unsigned)(uintptr_t)lds_dst;
  asm volatile("global_load_async_to_lds_b32 %0, %1, off"
               :: "v"(dst), "v"(gsrc)
               : "memory");
}
__device__ __forceinline__ void wait_async_le10() {
#if __has_builtin(__builtin_amdgcn_s_wait_asynccnt)
  __builtin_amdgcn_s_wait_asynccnt(10);
#else
  asm volatile("s_wait_asynccnt 10" ::: "memory");
#endif
}
__device__ __forceinline__ void wait_async_0() {
#if __has_builtin(__builtin_amdgcn_s_wait_asynccnt)
  __builtin_amdgcn_s_wait_asynccnt(0);
#else
  asm volatile("s_wait_asynccnt 0" ::: "memory");
#endif
}

// ---------------------------------------------------------------------------
// Kernel P0: initial state.  g1 = emb @ x[16:48], g2 = emb @ x[48:80],
//            f = x[80:144].  Grid (B, N/64), 256 threads.
// ---------------------------------------------------------------------------
__global__ void prep_gf(const float* __restrict__ x, const float* __restrict__ emb,
                        float* __restrict__ g1, float* __restrict__ g2,
                        float* __restrict__ f) {
  __shared__ float embl[EMB_ * NC_];
  int b = blockIdx.x, n0 = blockIdx.y * 64;
  const float* xb = x + (size_t)b * DIN_ * N_;
  for (int i = threadIdx.x; i < EMB_ * NC_; i += 256) embl[i] = emb[i];
  __syncthreads();
  int nl = threadIdx.x & 63, eg = threadIdx.x >> 6;  // 64 cols x 4 e-groups
  int n = n0 + nl;
  float a1[16], a2[16];
#pragma unroll
  for (int j = 0; j < 16; j++) { a1[j] = 0.f; a2[j] = 0.f; }
  for (int c = 0; c < NC_; c++) {
    float xv1 = xb[(size_t)(INSZ + c) * N_ + n];
    float xv2 = xb[(size_t)(INSZ + NC_ + c) * N_ + n];
#pragma unroll
    for (int j = 0; j < 16; j++) {
      float w = embl[(eg * 16 + j) * NC_ + c];
      a1[j] += w * xv1;
      a2[j] += w * xv2;
    }
  }
  size_t base = (size_t)b * EMB_ * N_;
#pragma unroll
  for (int j = 0; j < 16; j++) {
    int e = eg * 16 + j;
    g1[base + (size_t)e * N_ + n] = a1[j];
    g2[base + (size_t)e * N_ + n] = a2[j];
    f[base + (size_t)e * N_ + n] = xb[(size_t)(INSZ + 2 * NC_ + e) * N_ + n];
  }
}

// ---------------------------------------------------------------------------
// Kernel P1: head-1 softmax statistics.  Scores depend only on the constant
// s-band, so row-max m[q] and 1/Z[q] are computed ONCE for all 3 layers.
// Grid (B, N/256), 256 threads (one query each).
// ---------------------------------------------------------------------------
__global__ void prep_stats(const float* __restrict__ x, float* __restrict__ rowmax,
                           float* __restrict__ invz) {
  __shared__ float skl[INSZ * 16];
  int b = blockIdx.x, q = blockIdx.y * 256 + threadIdx.x;
  const float* xs = x + (size_t)b * DIN_ * N_;
  float sq[INSZ];
#pragma unroll
  for (int i = 0; i < INSZ; i++) sq[i] = xs[(size_t)i * N_ + q];
  const float cS = 1.0e10f / sqrtf(208.0f);  // LAM^2 / sqrt(D)
  float m = -INFINITY, z = 0.0f;
  for (int kc = 0; kc < MKV / 16; kc++) {
    __syncthreads();
    {
      int i = threadIdx.x >> 4, k = threadIdx.x & 15;
      skl[i * 16 + k] = xs[(size_t)i * N_ + kc * 16 + k];
    }
    __syncthreads();
#pragma unroll
    for (int k = 0; k < 16; k++) {
      float d = 0.f;
#pragma unroll
      for (int i = 0; i < INSZ; i++) d += sq[i] * skl[i * 16 + k];
      float l = cS * d;
      float nm = fmaxf(m, l);
      z = z * __expf(m - nm) + __expf(l - nm);
      m = nm;
    }
  }
  rowmax[(size_t)b * N_ + q] = m;
  invz[(size_t)b * N_ + q] = 1.0f / z;
}

// ---------------------------------------------------------------------------
// Kernel A (per layer): T = (kp/M)*diag(alpha_l)*(g1-g2)[:, :1008] @ s_kv^T
// (64x16 per batch).  Grid (B), 256 threads.
// ---------------------------------------------------------------------------
__global__ void head0_T(const float* __restrict__ x, const float* __restrict__ g1,
                        const float* __restrict__ g2,
                        const float* __restrict__ alpha_l,
                        const float* __restrict__ kparam, float* __restrict__ T) {
  __shared__ float dif[EMB_ * 16];
  __shared__ float sl[INSZ * 16];
  int b = blockIdx.x;
  const float* xs = x + (size_t)b * DIN_ * N_;
  const float* g1b = g1 + (size_t)b * EMB_ * N_;
  const float* g2b = g2 + (size_t)b * EMB_ * N_;
  int e = threadIdx.x & 63, ib = threadIdx.x >> 6;  // thread -> (e, 4 i's)
  float acc[4] = {0.f, 0.f, 0.f, 0.f};
  for (int kc = 0; kc < MKV / 16; kc++) {
    __syncthreads();
    for (int idx = threadIdx.x; idx < EMB_ * 16; idx += 256) {
      int ee = idx >> 4, nl = idx & 15;
      size_t o = (size_t)ee * N_ + kc * 16 + nl;
      dif[idx] = g1b[o] - g2b[o];
    }
    {
      int i = threadIdx.x >> 4, nl = threadIdx.x & 15;
      sl[i * 16 + nl] = xs[(size_t)i * N_ + kc * 16 + nl];
    }
    __syncthreads();
#pragma unroll
    for (int nl = 0; nl < 16; nl++) {
      float d = dif[e * 16 + nl];
#pragma unroll
      for (int j = 0; j < 4; j++) acc[j] += d * sl[(ib + 4 * j) * 16 + nl];
    }
  }
  float scale = alpha_l[e] * kparam[0] * (1.0f / (float)MKV);
#pragma unroll
  for (int j = 0; j < 4; j++)
    T[(size_t)b * EMB_ * INSZ + e * INSZ + (ib + 4 * j)] = acc[j] * scale;
}

// ---------------------------------------------------------------------------
// Kernel B (per layer): fused WMMA layer with double-buffered async staging.
// Per block: batch b, 64 query columns; 4 waves, 16 queries each.
//   1. head-1 attention  O_T[e,q] = sum_n g2[e,n] * p[n,q]  (63 key chunks;
//      K/V tiles staged via global_load_async_to_lds_b32, double buffered,
//      scores + P@V via v_wmma_f32_16x16x4_f32, softmax stats precomputed)
//   2. f += T @ s  (WMMA, K=16)
//   3. cross-attn: L = emb^T f_new (WMMA, K=64), softmax over 32 cats,
//      dg2 = emb @ p (WMMA, K=32)
//   4. g2_new = g2_old - O_T/M + dg2
// ---------------------------------------------------------------------------
__global__ void __launch_bounds__(128) layer_main(
    const float* __restrict__ x, const float* __restrict__ g2old,
    float* __restrict__ g2new, float* __restrict__ f,
    const float* __restrict__ Tmat, const float* __restrict__ rowmax,
    const float* __restrict__ invz, const float* __restrict__ emb) {
  __shared__ float skl[2][INSZ * 16];  // s key chunk   [i][key]  (dbl buf)
  __shared__ float vt[2][16 * EMB_];   // g2 key chunk, transposed [key][e]
  __shared__ float embl[EMB_ * NC_];   // emb  [e][c]
  __shared__ float embt[NC_ * EMB_];   // emb^T [c][e]
  __shared__ float tl[EMB_ * INSZ];    // T [e][i]
  __shared__ float pl[4][16 * 16];     // per-wave P tile [key][q]
  __shared__ float fl[4][EMB_ * 16];   // per-wave f_new  [e][q]
  __shared__ float pc[4][NC_ * 16];    // per-wave cross probs [c][q]

  int b = blockIdx.x;
  int tid = threadIdx.x;
  int w = tid >> 5, lane = tid & 31;
  int l15 = lane & 15;
  int half = (lane >> 4) << 1;  // K offset for this lane half (0 or 2)
  int hi8 = (lane >> 4) << 3;   // M offset for C/D rows (0 or 8)
  int q0 = blockIdx.y * 64 + w * 16;

  const float* xs = x + (size_t)b * DIN_ * N_;
  const float* g2b = g2old + (size_t)b * EMB_ * N_;
  float* g2nb = g2new + (size_t)b * EMB_ * N_;
  float* fb = f + (size_t)b * EMB_ * N_;
  const float* Tb = Tmat + (size_t)b * EMB_ * INSZ;

  for (int i = tid; i < EMB_ * NC_; i += 128) {
    float v = emb[i];
    embl[i] = v;
    embt[(i & (NC_ - 1)) * EMB_ + (i >> 5)] = v;  // i = e*32 + c
  }
  for (int i = tid; i < EMB_ * INSZ; i += 128) tl[i] = Tb[i];

  // B operand (queries): B[K=i][N=q], lane half selects K pair
  v2f bq[4];
#pragma unroll
  for (int kk = 0; kk < 4; kk++) {
    bq[kk].x = xs[(size_t)(4 * kk + half) * N_ + q0 + l15];
    bq[kk].y = xs[(size_t)(4 * kk + half + 1) * N_ + q0 + l15];
  }
  float mq = rowmax[(size_t)b * N_ + q0 + l15];
  float izq = invz[(size_t)b * N_ + q0 + l15];
  const float cS = 1.0e10f / sqrtf(208.0f);

  // Issue one staging round: 8 vt + 2 skl async b32 loads per thread
  // (10 ASYNCcnt increments per wave; completion is in-order).
  auto issue_stage = [&](int kc, int buf) {
    int n0k = kc * 16;
#pragma unroll
    for (int j = 0; j < 8; j++) {
      int idx = tid + j * 128;
      int ee = idx >> 4, k = idx & 15;
      async_ld_b32(&vt[buf][k * EMB_ + ee], g2b + (size_t)ee * N_ + n0k + k);
    }
#pragma unroll
    for (int j = 0; j < 2; j++) {
      int idx = tid + j * 128;
      int i_ = idx >> 4, k = idx & 15;
      async_ld_b32(&skl[buf][i_ * 16 + k], xs + (size_t)i_ * N_ + n0k + k);
    }
  };

  v8f accO[4];
#pragma unroll
  for (int t = 0; t < 4; t++) accO[t] = zero8();

  issue_stage(0, 0);
  for (int kc = 0; kc < MKV / 16; kc++) {
    int buf = kc & 1;
    __syncthreads();  // all waves done reading buf^1 from previous iteration
    if (kc + 1 < MKV / 16) {
      issue_stage(kc + 1, buf ^ 1);   // overlap next chunk with this compute
      wait_async_le10();              // current chunk's 10 loads have landed
      // pull a future chunk toward the WGP caches
      if (kc + 4 < MKV / 16)
        __builtin_prefetch(g2b + (size_t)(tid >> 4) * N_ + (kc + 4) * 16, 0, 3);
    } else {
      wait_async_0();
    }
    __syncthreads();  // every wave's async data visible in LDS

    // scores S_T[key,q] = sum_i s[i,key]*s[i,q]   (A: M=key, K=i)
    v8f sc = zero8();
#pragma unroll
    for (int kk = 0; kk < 4; kk++) {
      v2f a;
      a.x = skl[buf][(4 * kk + half) * 16 + l15];
      a.y = skl[buf][(4 * kk + half + 1) * 16 + l15];
      sc = wmma4(a, bq[kk], sc);
    }
    // p = exp(cS*s - max)*invZ;  C layout -> LDS [key][q]
    float* pw = pl[w];
#pragma unroll
    for (int r = 0; r < 8; r++)
      pw[(r + hi8) * 16 + l15] = __expf(cS * sc[r] - mq) * izq;
    // reload P in B layout (K=key, N=q)
    v2f bp[4];
#pragma unroll
    for (int kk = 0; kk < 4; kk++) {
      bp[kk].x = pw[(4 * kk + half) * 16 + l15];
      bp[kk].y = pw[(4 * kk + half + 1) * 16 + l15];
    }
    // O_T[e,q] += V[e,key] * P[key,q]   (A: M=e, K=key)
#pragma unroll
    for (int t = 0; t < 4; t++) {
#pragma unroll
      for (int kk = 0; kk < 4; kk++) {
        v2f a;
        a.x = vt[buf][(4 * kk + half) * EMB_ + 16 * t + l15];
        a.y = vt[buf][(4 * kk + half + 1) * EMB_ + 16 * t + l15];
        accO[t] = wmma4(a, bp[kk], accO[t]);
      }
    }
  }

  // f_new = f + T @ s   (A: M=e, K=i; B = bq again)
  float* fw = fl[w];
#pragma unroll
  for (int t = 0; t < 4; t++) {
    v8f a8 = zero8();
#pragma unroll
    for (int kk = 0; kk < 4; kk++) {
      v2f a;
      a.x = tl[(16 * t + l15) * INSZ + 4 * kk + half];
      a.y = tl[(16 * t + l15) * INSZ + 4 * kk + half + 1];
      a8 = wmma4(a, bq[kk], a8);
    }
#pragma unroll
    for (int r = 0; r < 8; r++) {
      int e = 16 * t + r + hi8;
      size_t o = (size_t)e * N_ + q0 + l15;
      float fn = fb[o] + a8[r];
      fb[o] = fn;               // in place: block owns these columns
      fw[e * 16 + l15] = fn;
    }
  }

  // cross-attn logits L[c,q] = sum_e emb[e,c]*f_new[e,q]  (A: M=c, K=e)
  v8f accL[2];
#pragma unroll
  for (int tc = 0; tc < 2; tc++) {
    v8f a8 = zero8();
#pragma unroll
    for (int ks = 0; ks < 16; ks++) {
      v2f a, bb;
      a.x = embl[(4 * ks + half) * NC_ + 16 * tc + l15];
      a.y = embl[(4 * ks + half + 1) * NC_ + 16 * tc + l15];
      bb.x = fw[(4 * ks + half) * 16 + l15];
      bb.y = fw[(4 * ks + half + 1) * 16 + l15];
      a8 = wmma4(a, bb, a8);
    }
    accL[tc] = a8;
  }
  // softmax over 32 categories: 16 in this lane, 16 in partner lane (xor 16)
  float mC = -INFINITY;
#pragma unroll
  for (int tc = 0; tc < 2; tc++)
#pragma unroll
    for (int r = 0; r < 8; r++) mC = fmaxf(mC, accL[tc][r]);
  mC = fmaxf(mC, __shfl_xor(mC, 16, 32));
  float sC = 0.f, pcv[2][8];
#pragma unroll
  for (int tc = 0; tc < 2; tc++)
#pragma unroll
    for (int r = 0; r < 8; r++) {
      float ev = __expf(accL[tc][r] - mC);
      pcv[tc][r] = ev;
      sC += ev;
    }
  sC += __shfl_xor(sC, 16, 32);
  float isC = 1.0f / sC;
  float* pcw = pc[w];
#pragma unroll
  for (int tc = 0; tc < 2; tc++)
#pragma unroll
    for (int r = 0; r < 8; r++)
      pcw[(16 * tc + r + hi8) * 16 + l15] = pcv[tc][r] * isC;

  // dg2[e,q] = emb[e,c] * p[c,q]   (A: M=e, K=c via emb^T)
  const float invM = 1.0f / (float)MKV;
#pragma unroll
  for (int t = 0; t < 4; t++) {
    v8f a8 = zero8();
#pragma unroll
    for (int kk = 0; kk < 8; kk++) {
      v2f a, bb;
      a.x = embt[(4 * kk + half) * EMB_ + 16 * t + l15];
      a.y = embt[(4 * kk + half + 1) * EMB_ + 16 * t + l15];
      bb.x = pcw[(4 * kk + half) * 16 + l15];
      bb.y = pcw[(4 * kk + half + 1) * 16 + l15];
      a8 = wmma4(a, bb, a8);
    }
#pragma unroll
    for (int r = 0; r < 8; r++) {
      int e = 16 * t + r + hi8;
      size_t o = (size_t)e * N_ + q0 + l15;
      g2nb[o] = g2b[o] - accO[t][r] * invM + a8[r];
    }
  }
}

// ---------------------------------------------------------------------------
// Kernel F: logits = f^T emb, predictions = softmax(logits).
// Grid (B, N/256), 256 threads (one column each).
// ---------------------------------------------------------------------------
__global__ void finalize(const float* __restrict__ f, const float* __restrict__ emb,
                         float* __restrict__ out) {
  __shared__ float embl[EMB_ * NC_];
  int b = blockIdx.x, n = blockIdx.y * 256 + threadIdx.x;
  for (int i = threadIdx.x; i < EMB_ * NC_; i += 256) embl[i] = emb[i];
  __syncthreads();
  const float* fb = f + (size_t)b * EMB_ * N_;
  float acc[NC_];
#pragma unroll
  for (int c = 0; c < NC_; c++) acc[c] = 0.f;
  for (int e = 0; e < EMB_; e++) {
    float fv = fb[(size_t)e * N_ + n];
#pragma unroll
    for (int c = 0; c < NC_; c++) acc[c] += fv * embl[e * NC_ + c];
  }
  float m = -INFINITY;
#pragma unroll
  for (int c = 0; c < NC_; c++) m = fmaxf(m, acc[c]);
  float s = 0.f, ex[NC_];
#pragma unroll
  for (int c = 0; c < NC_; c++) {
    ex[c] = __expf(acc[c] - m);
    s += ex[c];
  }
  float is = 1.0f / s;
  size_t base = ((size_t)b * N_ + n) * NC_;
  size_t poff = (size_t)B_ * N_ * NC_;
#pragma unroll
  for (int c = 0; c < NC_; c++) {
    out[base + c] = acc[c];
    out[poff + base + c] = ex[c] * is;
  }
}

// ---------------------------------------------------------------------------
extern "C" void kernel_launch(void* const* d_in, const int* in_sizes, int n_in,
                              void* d_out, int out_size, void* d_ws,
                              size_t ws_size, hipStream_t stream) {
  (void)in_sizes; (void)n_in; (void)out_size; (void)ws_size;
  const float* x = (const float*)d_in[0];       // (B, DIN, N)
  const float* alpha = (const float*)d_in[1];   // (3, 64)
  const float* kparam = (const float*)d_in[2];  // (1,)
  const float* emb = (const float*)d_in[3];     // (64, 32)

  // Workspace: g1, g2(a), g2(b), f  (each B*64*N f32) + T + softmax stats
  float* ws = (float*)d_ws;
  size_t gsz = (size_t)B_ * EMB_ * N_;
  float* g1 = ws;
  float* g2a = g1 + gsz;
  float* g2b = g2a + gsz;
  float* fbuf = g2b + gsz;
  float* Tm = fbuf + gsz;
  float* rmax = Tm + (size_t)B_ * EMB_ * INSZ;
  float* izv = rmax + (size_t)B_ * N_;   // total ~8.5M floats (~34 MB)

  prep_gf<<<dim3(B_, N_ / 64), 256, 0, stream>>>(x, emb, g1, g2a, fbuf);
  prep_stats<<<dim3(B_, N_ / 256), 256, 0, stream>>>(x, rmax, izv);

  float* cur = g2a;
  float* nxt = g2b;
  for (int l = 0; l < NLAYERS; l++) {
    head0_T<<<dim3(B_), 256, 0, stream>>>(x, g1, cur, alpha + l * EMB_, kparam, Tm);
    layer_main<<<dim3(B_, N_ / 64), 128, 0, stream>>>(x, cur, nxt, fbuf, Tm, rmax,
                                                      izv, emb);
    float* t = cur; cur = nxt; nxt = t;
  }
  finalize<<<dim3(B_, N_ / 256), 256, 0, stream>>>(fbuf, emb, (float*)d_out);
}